// LinearClassifierOld_87041807220926
// MI455X (gfx1250) — compile-verified
//
#include <hip/hip_runtime.h>

// CDNA5 / gfx1250: wave32, WMMA 16x16x4 fp32 path.
typedef float v2f __attribute__((ext_vector_type(2)));
typedef float v8f __attribute__((ext_vector_type(8)));

#define BATCH       1048576
#define KDIM        128
#define NDIM        64
#define ROWS_PER_WG 128          // 8 waves x 16 rows
#define THREADS     256
#define KPAD        132          // 128 + 4 dwords pad -> conflict-free LDS fragment reads

__launch_bounds__(THREADS, 2)
__global__ void linear_classifier_wmma_f32(const float* __restrict__ z,
                                           const float* __restrict__ W,
                                           const float* __restrict__ bias,
                                           float* __restrict__ out) {
    // W transposed into LDS: sW[c*KPAD + k] = W[k*NDIM + c].  64*132*4 = 33,792 B.
    __shared__ __attribute__((aligned(16))) float sW[NDIM * KPAD];

    const int tid    = threadIdx.x;
    const int wave   = tid >> 5;       // 0..7  (wave32)
    const int lane   = tid & 31;
    const int half   = lane >> 4;      // 0: K pair {4k,4k+1}; 1: K pair {4k+2,4k+3}
    const int l15    = lane & 15;
    const int wgRow0 = blockIdx.x * ROWS_PER_WG;

    // ---- Prefetch this WG's z slab (128 rows x 512 B = 1024 cache lines) while we
    //      stage W; lowers to global_prefetch_b8 on gfx1250. -------------------------
    {
        const char* zb = (const char*)(z + (size_t)wgRow0 * KDIM);
        #pragma unroll
        for (int j = 0; j < 4; ++j)
            __builtin_prefetch(zb + (size_t)(tid * 4 + j) * 64, 0, 3);
    }

    // ---- Cooperative transpose-stage of W (coalesced global b32 reads). ------------
    for (int i = tid; i < KDIM * NDIM; i += THREADS) {
        int k = i >> 6;                // 0..127   (W is row-major [K][N], N=64)
        int c = i & (NDIM - 1);        // 0..63
        sW[c * KPAD + k] = W[i];
    }
    __syncthreads();

    // ---- Per-wave 16(M) x 64(N) output slab; 4 accumulators of 16x16 f32. ----------
    const int rowBase = wgRow0 + wave * 16;

    // A fragment source: 32-bit A 16x4 layout — lanes 0-15: M=lane, K={4k,4k+1};
    // lanes 16-31: M=lane-16, K={4k+2,4k+3}.  8-byte aligned b64 loads.
    const float* zRow = z + (size_t)(rowBase + l15) * KDIM + 2 * half;

    // B fragment source (mirror of A across the K/N transpose): lanes 0-15: N=l15,
    // K={4k,4k+1}; lanes 16-31: N=l15, K={4k+2,4k+3}.  Conflict-free ds_load_b64:
    // bank = (4*l15 + k + 2*half) mod 64, all 32 lane accesses distinct.
    const float* sWc = sW + (size_t)l15 * KPAD + 2 * half;

    v8f acc0 = {}, acc1 = {}, acc2 = {}, acc3 = {};

    #pragma unroll 4
    for (int k = 0; k < KDIM; k += 4) {
        v2f a  = *(const v2f*)(zRow + k);
        v2f b0 = *(const v2f*)(sWc + k + 0 * 16 * KPAD);
        v2f b1 = *(const v2f*)(sWc + k + 1 * 16 * KPAD);
        v2f b2 = *(const v2f*)(sWc + k + 2 * 16 * KPAD);
        v2f b3 = *(const v2f*)(sWc + k + 3 * 16 * KPAD);

        // D = A(16x4) x B(4x16) + C, full fp32, RNE.
        acc0 = __builtin_amdgcn_wmma_f32_16x16x4_f32(false, a, false, b0, (short)0, acc0, false, false);
        acc1 = __builtin_amdgcn_wmma_f32_16x16x4_f32(false, a, false, b1, (short)0, acc1, false, false);
        acc2 = __builtin_amdgcn_wmma_f32_16x16x4_f32(false, a, false, b2, (short)0, acc2, false, false);
        acc3 = __builtin_amdgcn_wmma_f32_16x16x4_f32(false, a, false, b3, (short)0, acc3, false, false);
    }

    // ---- Bias + store.  C/D layout: VGPR v, lanes 0-15 -> M=v; lanes 16-31 -> M=8+v;
    //      N = lane&15 within each 16-col tile. -------------------------------------
    const float bv0 = bias[ 0 + l15];
    const float bv1 = bias[16 + l15];
    const float bv2 = bias[32 + l15];
    const float bv3 = bias[48 + l15];

    float* o = out + (size_t)(rowBase + 8 * half) * NDIM + l15;
    #pragma unroll
    for (int v = 0; v < 8; ++v) {
        float* orow = o + (size_t)v * NDIM;
        orow[ 0] = acc0[v] + bv0;
        orow[16] = acc1[v] + bv1;
        orow[32] = acc2[v] + bv2;
        orow[48] = acc3[v] + bv3;
    }
}

extern "C" void kernel_launch(void* const* d_in, const int* in_sizes, int n_in,
                              void* d_out, int out_size, void* d_ws, size_t ws_size,
                              hipStream_t stream) {
    const float* z    = (const float*)d_in[0];   // [BATCH, 128] f32
    const float* W    = (const float*)d_in[1];   // [128, 64]    f32
    const float* bias = (const float*)d_in[2];   // [64]         f32
    float*       out  = (float*)d_out;           // [BATCH, 64]  f32

    (void)in_sizes; (void)n_in; (void)out_size; (void)d_ws; (void)ws_size;

    linear_classifier_wmma_f32<<<BATCH / ROWS_PER_WG, THREADS, 0, stream>>>(z, W, bias, out);
}